// MemoryAugmentedModel_30322469109988
// MI455X (gfx1250) — compile-verified
//
#include <hip/hip_runtime.h>
#include <math.h>

// Problem constants (match reference)
#define N_NODES 2048
#define DIM     2048
#define HEADS   4
#define HPAD    16      // heads padded to WMMA N=16
#define RANK    32
#define BATCH   4
#define SEQ     2048
#define LORA_SCALE 2.0f
#define KSPLIT  4       // K-dim split for the WMMA a_src kernel
#define JSPLIT  8       // j-dim split for the weighted-sum kernel

typedef __attribute__((ext_vector_type(2))) float v2f;
typedef __attribute__((ext_vector_type(4))) float v4f;
typedef __attribute__((ext_vector_type(8))) float v8f;

// ---------------------------------------------------------------------------
// K1: stream gat_w once; produce
//   v_src_pad[h][k] = sum_d att_src[h,d] * gat_w[h*D+d, k]   (h<4; h=4..15 -> 0)
//   v_dst[h][k]     = sum_d att_dst[h,d] * gat_w[h*D+d, k]
// grid (DIM/256, HPAD), block 256
__global__ __launch_bounds__(256) void k1_compute_v(
    const float* __restrict__ gat_w, const float* __restrict__ att_src,
    const float* __restrict__ att_dst, float* __restrict__ v_src_pad,
    float* __restrict__ v_dst) {
  const int k = blockIdx.x * 256 + threadIdx.x;
  const int h = blockIdx.y;
  if (h >= HEADS) { v_src_pad[h * DIM + k] = 0.0f; return; }
  const float* wbase = gat_w + (size_t)h * DIM * DIM;
  float as = 0.0f, ad = 0.0f;
  #pragma unroll 4
  for (int d = 0; d < DIM; ++d) {
    float w = wbase[(size_t)d * DIM + k];
    as = fmaf(att_src[h * DIM + d], w, as);
    ad = fmaf(att_dst[h * DIM + d], w, ad);
  }
  v_src_pad[h * DIM + k] = as;
  v_dst[h * DIM + k] = ad;
}

// ---------------------------------------------------------------------------
// K2: a_src_part[kc][n][h] = sum_{k in chunk kc} nf[n,k] * v_src[h,k]
// via V_WMMA_F32_16X16X4_F32.  One wave per (16-node tile, K-chunk).
// A 16x4 layout: lanes 0-15 -> M=lane, VGPR0=K0,VGPR1=K1; lanes16-31 -> K2,K3.
// C layout: VGPR r: lanes0-15 (M=r,N=lane), lanes16-31 (M=8+r,N=lane-16).
__global__ __launch_bounds__(32) void k2_asrc_wmma(
    const float* __restrict__ nf, const float* __restrict__ v_src_pad,
    float* __restrict__ a_src_part /*[KSPLIT][N][HEADS]*/) {
  const int lane  = threadIdx.x;
  const int m0    = blockIdx.x * 16;
  const int kc    = blockIdx.y;
  const int kBase = kc * (DIM / KSPLIT);
  const int mrow  = m0 + (lane & 15);
  const int nCol  = lane & 15;
  const int kOff  = (lane < 16) ? 0 : 2;
  const float* aPtr = nf        + (size_t)mrow * DIM + kBase + kOff;
  const float* bPtr = v_src_pad + (size_t)nCol * DIM + kBase + kOff;
  v8f acc = {};
  for (int k = 0; k < DIM / KSPLIT; k += 4) {
    v2f a = *(const v2f*)(aPtr + k);   // 8B aligned: k%4==0, kOff in {0,2}
    v2f b = *(const v2f*)(bPtr + k);
    acc = __builtin_amdgcn_wmma_f32_16x16x4_f32(
        /*neg_a=*/false, a, /*neg_b=*/false, b,
        /*c_mod=*/(short)0, acc, /*reuse_a=*/false, /*reuse_b=*/false);
  }
  const int mBase = (lane < 16) ? 0 : 8;
  if (nCol < HEADS) {
    #pragma unroll
    for (int r = 0; r < 8; ++r)
      a_src_part[((size_t)kc * N_NODES + (m0 + mBase + r)) * HEADS + nCol] = acc[r];
  }
}

// K2b: a_src[i] = sum_kc a_src_part[kc][i],  i in [0, N*HEADS)
__global__ __launch_bounds__(256) void k2b_reduce(
    const float* __restrict__ part, float* __restrict__ a_src) {
  const int i = blockIdx.x * 256 + threadIdx.x;
  float s = 0.0f;
  #pragma unroll
  for (int kc = 0; kc < KSPLIT; ++kc) s += part[(size_t)kc * N_NODES * HEADS + i];
  a_src[i] = s;
}

// ---------------------------------------------------------------------------
// K3: per head: a_dst = nf[N-1,:]·v_dst[h,:]; softmax_j(leaky(a_dst+a_src[j,h]))
// grid HEADS blocks, 256 threads
__global__ __launch_bounds__(256) void k3_softmax(
    const float* __restrict__ nf, const float* __restrict__ v_dst,
    const float* __restrict__ a_src, float* __restrict__ attn /*[H][N]*/) {
  __shared__ float red[256];
  const int h = blockIdx.x, tid = threadIdx.x;
  const float* nfl = nf + (size_t)(N_NODES - 1) * DIM;
  float acc = 0.0f;
  for (int k = tid; k < DIM; k += 256) acc = fmaf(nfl[k], v_dst[h * DIM + k], acc);
  red[tid] = acc; __syncthreads();
  for (int s = 128; s > 0; s >>= 1) { if (tid < s) red[tid] += red[tid + s]; __syncthreads(); }
  const float adst = red[0]; __syncthreads();

  float mx = -3.4e38f;
  for (int j = tid; j < N_NODES; j += 256) {
    float l = adst + a_src[(size_t)j * HEADS + h];
    l = fmaxf(l, 0.2f * l);                       // leaky_relu(0.2)
    mx = fmaxf(mx, l);
  }
  red[tid] = mx; __syncthreads();
  for (int s = 128; s > 0; s >>= 1) { if (tid < s) red[tid] = fmaxf(red[tid], red[tid + s]); __syncthreads(); }
  const float gmax = red[0]; __syncthreads();

  float sum = 0.0f;
  for (int j = tid; j < N_NODES; j += 256) {
    float l = adst + a_src[(size_t)j * HEADS + h];
    l = fmaxf(l, 0.2f * l);
    float e = __expf(l - gmax);
    attn[(size_t)h * N_NODES + j] = e;
    sum += e;
  }
  red[tid] = sum; __syncthreads();
  for (int s = 128; s > 0; s >>= 1) { if (tid < s) red[tid] += red[tid + s]; __syncthreads(); }
  const float inv = 1.0f / red[0];
  for (int j = tid; j < N_NODES; j += 256)
    attn[(size_t)h * N_NODES + j] *= inv;
}

// ---------------------------------------------------------------------------
// K4: m_part[jc][h][k] = sum_{j in chunk jc} attn[h,j]*nf[j,k]
// grid (DIM/256, JSPLIT), block 256 -> 512 waves, 256-deep chains
__global__ __launch_bounds__(256) void k4_weighted_sum(
    const float* __restrict__ nf, const float* __restrict__ attn,
    float* __restrict__ m_part /*[JSPLIT][H][D]*/) {
  const int k  = blockIdx.x * 256 + threadIdx.x;
  const int jc = blockIdx.y;
  const int j0 = jc * (N_NODES / JSPLIT);
  const int j1 = j0 + (N_NODES / JSPLIT);
  float a0 = 0, a1 = 0, a2 = 0, a3 = 0;
  for (int j = j0; j < j1; ++j) {
    float f = nf[(size_t)j * DIM + k];
    a0 = fmaf(attn[0 * N_NODES + j], f, a0);
    a1 = fmaf(attn[1 * N_NODES + j], f, a1);
    a2 = fmaf(attn[2 * N_NODES + j], f, a2);
    a3 = fmaf(attn[3 * N_NODES + j], f, a3);
  }
  float* mp = m_part + (size_t)jc * HEADS * DIM;
  mp[0 * DIM + k] = a0; mp[1 * DIM + k] = a1;
  mp[2 * DIM + k] = a2; mp[3 * DIM + k] = a3;
}

// K4b: m[i] = sum_jc m_part[jc][i],  i in [0, H*DIM)
__global__ __launch_bounds__(256) void k4b_reduce(
    const float* __restrict__ m_part, float* __restrict__ m) {
  const int i = blockIdx.x * 256 + threadIdx.x;
  float s = 0.0f;
  #pragma unroll
  for (int jc = 0; jc < JSPLIT; ++jc) s += m_part[(size_t)jc * HEADS * DIM + i];
  m[i] = s;
}

// ---------------------------------------------------------------------------
// K5: out_last[d] = (1/H)*sum_h m[h,:]·gat_w[h*D+d,:] + bias[d]
// grid DIM blocks, 256 threads (second gat_w pass; should be L2-resident)
__global__ __launch_bounds__(256) void k5_out_last(
    const float* __restrict__ gat_w, const float* __restrict__ m,
    const float* __restrict__ gat_bias, float* __restrict__ out_last) {
  __shared__ float red[256];
  const int d = blockIdx.x, tid = threadIdx.x;
  float acc = 0.0f;
  #pragma unroll
  for (int h = 0; h < HEADS; ++h) {
    const float* row = gat_w + (size_t)(h * DIM + d) * DIM;
    const float* mh  = m + (size_t)h * DIM;
    for (int k = tid; k < DIM; k += 256) acc = fmaf(row[k], mh[k], acc);
  }
  red[tid] = acc; __syncthreads();
  for (int s = 128; s > 0; s >>= 1) { if (tid < s) red[tid] += red[tid + s]; __syncthreads(); }
  if (tid == 0) out_last[d] = red[0] * (1.0f / HEADS) + gat_bias[d];
}

// ---------------------------------------------------------------------------
// K6: LayerNorm over out_last -> mem.  1 block, 256 threads
__global__ __launch_bounds__(256) void k6_layernorm(
    const float* __restrict__ out_last, const float* __restrict__ gamma,
    const float* __restrict__ beta, float* __restrict__ mem) {
  __shared__ float rs[256], rq[256];
  const int tid = threadIdx.x;
  float s = 0, q = 0;
  for (int d = tid; d < DIM; d += 256) { float x = out_last[d]; s += x; q = fmaf(x, x, q); }
  rs[tid] = s; rq[tid] = q; __syncthreads();
  for (int t = 128; t > 0; t >>= 1) {
    if (tid < t) { rs[tid] += rs[tid + t]; rq[tid] += rq[tid + t]; }
    __syncthreads();
  }
  const float mu  = rs[0] * (1.0f / DIM);
  const float var = rq[0] * (1.0f / DIM) - mu * mu;
  const float rinv = rsqrtf(var + 1e-5f);
  for (int d = tid; d < DIM; d += 256)
    mem[d] = (out_last[d] - mu) * rinv * gamma[d] + beta[d];
}

// ---------------------------------------------------------------------------
// K7: t[r] = mem · lora_a[r,:].  grid RANK blocks, 256 threads
__global__ __launch_bounds__(256) void k7_lora_t(
    const float* __restrict__ mem, const float* __restrict__ lora_a,
    float* __restrict__ t) {
  __shared__ float red[256];
  const int r = blockIdx.x, tid = threadIdx.x;
  float acc = 0.0f;
  for (int k = tid; k < DIM; k += 256) acc = fmaf(mem[k], lora_a[(size_t)r * DIM + k], acc);
  red[tid] = acc; __syncthreads();
  for (int s = 128; s > 0; s >>= 1) { if (tid < s) red[tid] += red[tid + s]; __syncthreads(); }
  if (tid == 0) t[r] = red[0];
}

// ---------------------------------------------------------------------------
// K8: offset[d] = mem·proj_w[d,:] + proj_b[d] + 2*sum_r lora_b[d,r]*t[r]
// grid DIM blocks, 256 threads
__global__ __launch_bounds__(256) void k8_offset(
    const float* __restrict__ mem, const float* __restrict__ proj_w,
    const float* __restrict__ proj_b, const float* __restrict__ lora_b,
    const float* __restrict__ t, float* __restrict__ offset) {
  __shared__ float red[256];
  const int d = blockIdx.x, tid = threadIdx.x;
  float acc = 0.0f;
  for (int k = tid; k < DIM; k += 256) acc = fmaf(mem[k], proj_w[(size_t)d * DIM + k], acc);
  red[tid] = acc; __syncthreads();
  for (int s = 128; s > 0; s >>= 1) { if (tid < s) red[tid] += red[tid + s]; __syncthreads(); }
  if (tid == 0) {
    float lora = 0.0f;
    #pragma unroll
    for (int r = 0; r < RANK; ++r) lora = fmaf(lora_b[(size_t)d * RANK + r], t[r], lora);
    offset[d] = red[0] + proj_b[d] + LORA_SCALE * lora;
  }
}

// ---------------------------------------------------------------------------
// K9: embeds[b,s,:] = embed[ids[b,s],:] (+ offset if s==0).  Dominant kernel:
// ~134 MB HBM traffic. Vectorized with clang ext-vector float4, prefetch, and
// non-temporal stores so the write stream doesn't evict L2-resident data.
__global__ __launch_bounds__(256) void k9_gather(
    const float* __restrict__ embed, const int* __restrict__ ids,
    const float* __restrict__ offset, float* __restrict__ out) {
  const int bs  = blockIdx.x;            // 0 .. B*S-1
  const int s   = bs & (SEQ - 1);
  const int tok = ids[bs];
  const v4f* src = (const v4f*)(embed + (size_t)tok * DIM);
  const v4f* off = (const v4f*)offset;
  v4f*       dst = (v4f*)(out + (size_t)bs * DIM);
  const int tid = threadIdx.x;
  __builtin_prefetch(src + tid, 0, 1);   // global_prefetch into L2
  if (s == 0) {                          // uniform per block
    for (int i = tid; i < DIM / 4; i += 256) {
      v4f v = src[i] + off[i];
      __builtin_nontemporal_store(v, &dst[i]);
    }
  } else {
    for (int i = tid; i < DIM / 4; i += 256) {
      v4f v = src[i];
      __builtin_nontemporal_store(v, &dst[i]);
    }
  }
}

// ---------------------------------------------------------------------------
extern "C" void kernel_launch(void* const* d_in, const int* in_sizes, int n_in,
                              void* d_out, int out_size, void* d_ws, size_t ws_size,
                              hipStream_t stream) {
  const float* nf       = (const float*)d_in[0];
  const int*   ids      = (const int*)  d_in[1];
  const float* gat_w    = (const float*)d_in[2];
  const float* att_src  = (const float*)d_in[3];
  const float* att_dst  = (const float*)d_in[4];
  const float* gat_bias = (const float*)d_in[5];
  const float* ln_gamma = (const float*)d_in[6];
  const float* ln_beta  = (const float*)d_in[7];
  const float* proj_w   = (const float*)d_in[8];
  const float* proj_b   = (const float*)d_in[9];
  const float* lora_a   = (const float*)d_in[10];
  const float* lora_b   = (const float*)d_in[11];
  const float* embed    = (const float*)d_in[12];
  float* out = (float*)d_out;

  // workspace carve-up (floats)
  float* ws         = (float*)d_ws;
  float* v_src_pad  = ws;                                   // HPAD*DIM        = 32768
  float* v_dst      = v_src_pad + HPAD * DIM;               // HEADS*DIM       =  8192
  float* a_src_part = v_dst + HEADS * DIM;                  // KSPLIT*N*HEADS  = 32768
  float* a_src      = a_src_part + KSPLIT * N_NODES * HEADS;// N*HEADS         =  8192
  float* attn       = a_src + N_NODES * HEADS;              // HEADS*N         =  8192
  float* m_part     = attn + HEADS * N_NODES;               // JSPLIT*HEADS*DIM= 65536
  float* m          = m_part + JSPLIT * HEADS * DIM;        // HEADS*DIM       =  8192
  float* out_last   = m + HEADS * DIM;                      // DIM             =  2048
  float* mem        = out_last + DIM;                       // DIM             =  2048
  float* t          = mem + DIM;                            // RANK            =    32
  float* offset     = t + RANK;                             // DIM             =  2048

  k1_compute_v   <<<dim3(DIM / 256, HPAD), 256, 0, stream>>>(gat_w, att_src, att_dst, v_src_pad, v_dst);
  k2_asrc_wmma   <<<dim3(N_NODES / 16, KSPLIT), 32, 0, stream>>>(nf, v_src_pad, a_src_part);
  k2b_reduce     <<<(N_NODES * HEADS) / 256, 256, 0, stream>>>(a_src_part, a_src);
  k3_softmax     <<<HEADS, 256, 0, stream>>>(nf, v_dst, a_src, attn);
  k4_weighted_sum<<<dim3(DIM / 256, JSPLIT), 256, 0, stream>>>(nf, attn, m_part);
  k4b_reduce     <<<(HEADS * DIM) / 256, 256, 0, stream>>>(m_part, m);
  k5_out_last    <<<DIM, 256, 0, stream>>>(gat_w, m, gat_bias, out_last);
  k6_layernorm   <<<1, 256, 0, stream>>>(out_last, ln_gamma, ln_beta, mem);
  k7_lora_t      <<<RANK, 256, 0, stream>>>(mem, lora_a, t);
  k8_offset      <<<DIM, 256, 0, stream>>>(mem, proj_w, proj_b, lora_b, t, offset);
  k9_gather      <<<BATCH * SEQ, 256, 0, stream>>>(embed, ids, offset, out);
}